// MAFF_90357521973298
// MI455X (gfx1250) — compile-verified
//
#include <hip/hip_runtime.h>
#include <hip/hip_bf16.h>

// ---------------------------------------------------------------------------
// LoFTR-style matcher for MI455X (gfx1250, wave32).
// Heavy op: batched f32 GEMM (2 x 4096x4096x256) -> V_WMMA_F32_16X16X4_F32,
// staged global->LDS via the Tensor Data Mover (double buffered, TENSORcnt).
// ---------------------------------------------------------------------------

typedef __attribute__((ext_vector_type(2))) float        v2f;
typedef __attribute__((ext_vector_type(8))) float        v8f;
typedef __attribute__((ext_vector_type(4))) unsigned int v4u;
typedef __attribute__((ext_vector_type(8))) int          v8i;
typedef __attribute__((ext_vector_type(4))) int          v4i;

#if defined(__has_builtin)
#if __has_builtin(__builtin_amdgcn_tensor_load_to_lds) && \
    __has_builtin(__builtin_amdgcn_s_wait_tensorcnt)
#define MAFF_HAVE_TDM 1
#endif
#endif
#ifndef MAFF_HAVE_TDM
#define MAFF_HAVE_TDM 0
#endif

namespace maff {
constexpr int B_    = 2;
constexpr int L0_   = 4096;
constexpr int L1_   = 4096;
constexpr int CH_   = 256;
constexpr int H0_   = 64;   // = W0 = H1 = W1
constexpr int WS_   = 5;
constexpr int MMAX_ = 1024;
constexpr int KCH_  = 64;   // K chunk staged per LDS buffer
constexpr int LDSW_ = 68;   // padded LDS row stride (floats) -> bank spread
constexpr float SCALE_    = 8.0f;                     // 512/64
constexpr float SIMSCALE_ = 1.0f / (256.0f * 0.1f);   // (1/C)/TEMP
constexpr float INVSQRTC_ = 0.0625f;                  // 1/sqrt(256)
}

// --- order-preserving float <-> uint key (for atomicMax over floats) -------
__device__ __forceinline__ unsigned int f2key(float f) {
  unsigned int b = __float_as_uint(f);
  return (b & 0x80000000u) ? ~b : (b | 0x80000000u);
}
__device__ __forceinline__ float key2f(unsigned int k) {
  unsigned int b = (k & 0x80000000u) ? (k ^ 0x80000000u) : ~k;
  return __uint_as_float(b);
}

#if MAFF_HAVE_TDM
// ---------------------------------------------------------------------------
// Issue one TDM descriptor: 64 rows x 64 channels f32 tile, tensor row stride
// 256 floats, LDS padding 4 DWORDs after every 64 DWORDs (-> 68-float rows).
// ---------------------------------------------------------------------------
__device__ __forceinline__ void maff_tdm_load(unsigned lds_off, unsigned long long ga) {
  using namespace maff;
  const v4u g0 = {
      1u,                                  // count=1, user descriptor
      lds_off,                             // lds_addr  [63:32]
      (unsigned)(ga & 0xFFFFFFFFu),        // global_addr low
      (unsigned)((ga >> 32) & 0x01FFFFFFu) // global_addr [56:32]
          | (2u << 30)                     // type=2 ("image")
  };
  const v8i g1 = {
      (int)0x07520000,  // data_size=4B, pad_enable, pad_interval=64dw, pad_amount=4dw
      (int)0x01000000,  // tensor_dim0 = 256 (bits 79:48, low16 here)
      (int)0x10000000,  // tensor_dim0 hi=0 | tensor_dim1 = 4096 (low16)
      (int)0x00400000,  // tensor_dim1 hi=0 | tile_dim0 = 64
      64,               // tile_dim1 = 64, tile_dim2 = 0
      256,              // tensor_dim0_stride = 256 (low32)
      0,                // stride hi / tensor_dim1_stride
      0
  };
  const v4i gz = {0, 0, 0, 0};
#if __clang_major__ >= 23
  const v8i gz8 = {0, 0, 0, 0, 0, 0, 0, 0};
  __builtin_amdgcn_tensor_load_to_lds(g0, g1, gz, gz, gz8, 0);
#else
  __builtin_amdgcn_tensor_load_to_lds(g0, g1, gz, gz, 0);
#endif
}
#endif

// ---------------------------------------------------------------------------
__global__ void MAFF_init_kernel(unsigned int* maxkey, int* cnt) {
  if (threadIdx.x == 0) { *maxkey = 0u; *cnt = 0; }
}

// ---------------------------------------------------------------------------
// Kernel 1: sim[b,i,j] = dot(feat0[b,i], feat1[b,j]) / (C*TEMP), masked,
// plus global max. 64x64 block tile, 4 waves, each wave a 32x32 register
// block (2x2 WMMA tiles -> 1 LDS load per WMMA). K staged in 64-chunks,
// double buffered via TDM (TENSORcnt) when available.
// ---------------------------------------------------------------------------
__global__ __launch_bounds__(128) void MAFF_sim_gemm(
    const float* __restrict__ f0, const float* __restrict__ f1,
    const unsigned char* __restrict__ mk0, const unsigned char* __restrict__ mk1,
    float* __restrict__ sim, unsigned int* __restrict__ maxkey) {
  using namespace maff;
  __shared__ float lds[2][2][64 * LDSW_];   // [buf][A|B][row*68 + k]

  const int bIdx = blockIdx.z;
  const int i0   = blockIdx.y * 64;
  const int j0   = blockIdx.x * 64;
  const int tid  = threadIdx.x;
  const int wave = tid >> 5;
  const int lane = tid & 31;
  const int half = lane >> 4;       // K-pair selector in WMMA frag layout
  const int l16  = lane & 15;
  const int mBase = (wave >> 1) * 32;
  const int nBase = (wave & 1) * 32;

  const float* A  = f0 + ((size_t)bIdx * L0_ + i0) * CH_;
  const float* Bp = f1 + ((size_t)bIdx * L1_ + j0) * CH_;

  v8f acc00 = {}, acc01 = {}, acc10 = {}, acc11 = {};

  const int NCHUNK = CH_ / KCH_;   // 4

#if MAFF_HAVE_TDM
  const unsigned long long aBase = (unsigned long long)(uintptr_t)A;
  const unsigned long long bBase = (unsigned long long)(uintptr_t)Bp;
  if (wave == 0) {  // wave-uniform branch; TDM issues once per wave
    maff_tdm_load((unsigned)(uintptr_t)&lds[0][0][0], aBase);
    maff_tdm_load((unsigned)(uintptr_t)&lds[0][1][0], bBase);
  }
#endif

  for (int k = 0; k < NCHUNK; ++k) {
    const int buf = k & 1;
#if MAFF_HAVE_TDM
    if (wave == 0) {
      if (k + 1 < NCHUNK) {  // prefetch next chunk into the other buffer
        const unsigned long long koff = (unsigned long long)(k + 1) * KCH_ * 4ull;
        maff_tdm_load((unsigned)(uintptr_t)&lds[buf ^ 1][0][0], aBase + koff);
        maff_tdm_load((unsigned)(uintptr_t)&lds[buf ^ 1][1][0], bBase + koff);
        // in-order TENSORcnt: <=2 outstanding -> chunk k's 2 loads are done
        __builtin_amdgcn_s_wait_tensorcnt(2);
      } else {
        __builtin_amdgcn_s_wait_tensorcnt(0);
      }
    }
    __syncthreads();
#else
    // Fallback: cooperative staging (float4, coalesced)
    for (int v = tid; v < 64 * 16; v += 128) {
      const int row = v >> 4;
      const int cg  = (v & 15) << 2;
      *(float4*)&lds[buf][0][row * LDSW_ + cg] =
          *(const float4*)(A + row * CH_ + k * KCH_ + cg);
      *(float4*)&lds[buf][1][row * LDSW_ + cg] =
          *(const float4*)(Bp + row * CH_ + k * KCH_ + cg);
    }
    __syncthreads();
#endif

    const float* As = &lds[buf][0][0];
    const float* Bs = &lds[buf][1][0];
#pragma unroll
    for (int kk = 0; kk < KCH_; kk += 4) {
      const int ko = kk + 2 * half;
      // A frag 16x4: lane -> M = lane%16, K = 2*(lane/16) + {0,1}; B symmetric.
      const v2f a0 = *(const v2f*)&As[(mBase + l16) * LDSW_ + ko];
      const v2f a1 = *(const v2f*)&As[(mBase + 16 + l16) * LDSW_ + ko];
      const v2f b0 = *(const v2f*)&Bs[(nBase + l16) * LDSW_ + ko];
      const v2f b1 = *(const v2f*)&Bs[(nBase + 16 + l16) * LDSW_ + ko];
      acc00 = __builtin_amdgcn_wmma_f32_16x16x4_f32(false, a0, false, b0, (short)0, acc00, false, false);
      acc01 = __builtin_amdgcn_wmma_f32_16x16x4_f32(false, a0, false, b1, (short)0, acc01, false, false);
      acc10 = __builtin_amdgcn_wmma_f32_16x16x4_f32(false, a1, false, b0, (short)0, acc10, false, false);
      acc11 = __builtin_amdgcn_wmma_f32_16x16x4_f32(false, a1, false, b1, (short)0, acc11, false, false);
    }
    __syncthreads();
  }

  // epilogue: scale, mask, store, track global max
  float lmax = -3.0e38f;
  auto emit = [&](const v8f& a, int mo, int no) {
    const int  col  = j0 + no + l16;
    const bool cmok = mk1[(size_t)bIdx * L1_ + col] != 0;
#pragma unroll
    for (int r = 0; r < 8; ++r) {
      const int row = i0 + mo + r + 8 * half;   // C/D layout: M = r + 8*(lane/16)
      float v = a[r] * SIMSCALE_;
      const bool ok = cmok && (mk0[(size_t)bIdx * L0_ + row] != 0);
      v = ok ? v : -1.0e9f;
      sim[((size_t)bIdx * L0_ + row) * L1_ + col] = v;
      lmax = fmaxf(lmax, v);
    }
  };
  emit(acc00, mBase,      nBase);
  emit(acc01, mBase,      nBase + 16);
  emit(acc10, mBase + 16, nBase);
  emit(acc11, mBase + 16, nBase + 16);

#pragma unroll
  for (int off = 16; off >= 1; off >>= 1)
    lmax = fmaxf(lmax, __shfl_xor(lmax, off, 32));
  if (lane == 0) atomicMax(maxkey, f2key(lmax));
}

// ---------------------------------------------------------------------------
// Kernel 2: cond = sim > 0.8*max  -> append (b,i,j) match triples (<= 1024)
// ---------------------------------------------------------------------------
__global__ __launch_bounds__(256) void MAFF_select(
    const float* __restrict__ sim, const unsigned int* __restrict__ maxkey,
    int* __restrict__ cnt, int* __restrict__ matches) {
  using namespace maff;
  const size_t total = (size_t)B_ * L0_ * L1_;
  const size_t i = (size_t)blockIdx.x * 256 + threadIdx.x;
  if (i >= total) return;
  const float thr = 0.8f * key2f(*maxkey);
  const float v = sim[i];
  if (v > thr) {
    const int pos = atomicAdd(cnt, 1);
    if (pos < MMAX_) {
      const int b   = (int)(i / ((size_t)L0_ * L1_));
      const int rem = (int)(i - (size_t)b * L0_ * L1_);
      matches[pos * 3 + 0] = b;
      matches[pos * 3 + 1] = rem / L1_;   // i index
      matches[pos * 3 + 2] = rem % L1_;   // j index
    }
  }
}

// ---------------------------------------------------------------------------
// Kernel 3: fine refinement. One wave per match slot; 25 active lanes own
// one 5x5 window cell each; softmax over rows (axis=1) via lane shuffles.
// ---------------------------------------------------------------------------
__global__ __launch_bounds__(32) void MAFF_fine(
    const float* __restrict__ f0, const float* __restrict__ f1,
    const int* __restrict__ cnt, const int* __restrict__ matches,
    float* __restrict__ fc0, float* __restrict__ fc1,
    float* __restrict__ stdo, float* __restrict__ valido) {
  using namespace maff;
  const int m    = blockIdx.x;
  const int lane = threadIdx.x;
  const int total = *cnt;
  const int nstored = total < MMAX_ ? total : MMAX_;

  if (m >= nstored) {  // invalid slot: reference zeroes everything via vf
    if (lane == 0) {
      fc0[2 * m] = 0.0f; fc0[2 * m + 1] = 0.0f;
      fc1[2 * m] = 0.0f; fc1[2 * m + 1] = 0.0f;
      stdo[m] = 0.0f;
      valido[m] = (m < total) ? 1.0f : 0.0f;
    }
    return;
  }

  const int b  = matches[3 * m + 0];
  const int ii = matches[3 * m + 1];
  const int jj = matches[3 * m + 2];

  const int h = lane / 5, w = lane % 5;          // lanes 25..31 idle
  const bool act = lane < 25;
  const int rr = jj / H0_ + h - (WS_ / 2);
  const int cc = jj % H0_ + w - (WS_ / 2);
  const bool inb = act && rr >= 0 && rr < H0_ && cc >= 0 && cc < H0_;

  const float* fp0 = f0 + ((size_t)b * L0_ + ii) * CH_;
  const float* fw  = f1 + ((size_t)b * L1_ + rr * H0_ + cc) * CH_;

  float acc = 0.0f;
  for (int c = 0; c < CH_; ++c) {
    const float a  = fp0[c];                 // uniform address -> broadcast
    const float bv = inb ? fw[c] : 0.0f;     // zero padding
    acc = fmaf(a, bv, acc);
  }
  float heat = acc * INVSQRTC_;
  if (!act) heat = -3.0e38f;

  // softmax over h (axis=1) per column w: gather lanes {w, 5+w, ..., 20+w}
  float mmax = -3.0e38f;
#pragma unroll
  for (int r = 0; r < WS_; ++r)
    mmax = fmaxf(mmax, __shfl(heat, r * WS_ + w, 32));
  const float e = act ? __expf(heat - mmax) : 0.0f;
  float s = 0.0f;
#pragma unroll
  for (int r = 0; r < WS_; ++r)
    s += __shfl(e, r * WS_ + w, 32);
  const float p = act ? (e / s) : 0.0f;

  // expectation / variance over grid = ((w-2)/2, (h-2)/2)
  const float gx = (float)(w - 2) * 0.5f;
  const float gy = (float)(h - 2) * 0.5f;
  float px = p * gx, py = p * gy;
  float px2 = p * gx * gx, py2 = p * gy * gy;
#pragma unroll
  for (int off = 16; off >= 1; off >>= 1) {
    px  += __shfl_xor(px,  off, 32);
    py  += __shfl_xor(py,  off, 32);
    px2 += __shfl_xor(px2, off, 32);
    py2 += __shfl_xor(py2, off, 32);
  }

  if (lane == 0) {
    const float cx = px, cy = py;
    const float vx = fmaxf(px2 - cx * cx, 1e-10f);
    const float vy = fmaxf(py2 - cy * cy, 1e-10f);
    const float st = sqrtf(vx) + sqrtf(vy);
    fc0[2 * m]     = (float)(ii % H0_) * SCALE_;
    fc0[2 * m + 1] = (float)(ii / H0_) * SCALE_;
    fc1[2 * m]     = cx * (float)(WS_ / 2) * SCALE_ + (float)(jj % H0_) * SCALE_;
    fc1[2 * m + 1] = cy * (float)(WS_ / 2) * SCALE_ + (float)(jj / H0_) * SCALE_;
    stdo[m]   = st;
    valido[m] = 1.0f;
  }
}

// ---------------------------------------------------------------------------
extern "C" void kernel_launch(void* const* d_in, const int* in_sizes, int n_in,
                              void* d_out, int out_size, void* d_ws, size_t ws_size,
                              hipStream_t stream) {
  using namespace maff;
  const float* f0 = (const float*)d_in[0];
  const float* f1 = (const float*)d_in[1];
  const unsigned char* m0 = (const unsigned char*)d_in[2];
  const unsigned char* m1 = (const unsigned char*)d_in[3];

  float* sim    = (float*)d_out;
  float* fc0    = sim + (size_t)B_ * L0_ * L1_;
  float* fc1    = fc0 + 2 * MMAX_;
  float* stdo   = fc1 + 2 * MMAX_;
  float* valido = stdo + MMAX_;

  unsigned int* maxkey = (unsigned int*)d_ws;
  int* cnt     = (int*)d_ws + 1;
  int* matches = (int*)d_ws + 4;   // 3*1024 ints

  MAFF_init_kernel<<<1, 32, 0, stream>>>(maxkey, cnt);

  dim3 g1(L1_ / 64, L0_ / 64, B_);
  MAFF_sim_gemm<<<g1, 128, 0, stream>>>(f0, f1, m0, m1, sim, maxkey);

  const size_t total = (size_t)B_ * L0_ * L1_;
  MAFF_select<<<(unsigned)((total + 255) / 256), 256, 0, stream>>>(sim, maxkey, cnt, matches);

  MAFF_fine<<<MMAX_, 32, 0, stream>>>(f0, f1, cnt, matches, fc0, fc1, stdo, valido);
}